// FraudDetectionNet_65481071397542
// MI455X (gfx1250) — compile-verified
//
#include <hip/hip_runtime.h>
#include <math.h>

typedef __attribute__((ext_vector_type(2))) float v2f;
typedef __attribute__((ext_vector_type(8))) float v8f;

#define ZC_ROWS 208  // 13 tiles * 16 patches, rows 196..207 zero-padded

// ---------------------------------------------------------------------------
// Kernel A: zc[p][k] = sum_w Z[k][w] * w_cls[1 + 4p + w],  Z[k][w] = 1-2*bit.
// Padded to 208 rows (zeros past patch 195) so the main kernel needs NO
// divergent guards around its zc loads (keeps EXEC uniform, kills the
// save/restore-exec SALU storm seen in round 1).
// ---------------------------------------------------------------------------
__global__ void zc_setup_kernel(const float* __restrict__ w_cls,
                                float* __restrict__ zc) {
  for (int i = threadIdx.x; i < ZC_ROWS * 16; i += blockDim.x) {
    int p = i >> 4;
    int k = i & 15;
    float acc = 0.f;
    if (p < 196) {
#pragma unroll
      for (int w = 0; w < 4; ++w) {
        float z = ((k >> (3 - w)) & 1) ? -1.f : 1.f;
        acc += z * w_cls[1 + 4 * p + w];
      }
    }
    zc[i] = acc;
  }
}

// ---------------------------------------------------------------------------
// Kernel B: one workgroup (128 threads = 4 waves) per image.
//  Phase 1: stage 28x28 image in LDS via float4, block-reduce sum/sumsq,
//           thread 0 runs the photonic MLP.
//  Phase 2: 13 tiles x 16 patches; per tile: build psi in WMMA A-layout,
//           8x v_wmma_f32_16x16x4_f32 (re & im, K=16 as 4 chunks),
//           accumulate (re^2+im^2)*zc per lane (unconditional loads).
//  Phase 3: block reduce, add classical term + bias, sigmoid, store.
// ---------------------------------------------------------------------------
__global__ __launch_bounds__(128) void fraud_fused_kernel(
    const float* __restrict__ x, const float* __restrict__ w_in,
    const float* __restrict__ b_in, const float* __restrict__ scale_in,
    const float* __restrict__ shift_in, const float* __restrict__ Wc,
    const float* __restrict__ bc, const float* __restrict__ scalec,
    const float* __restrict__ shiftc, const float* __restrict__ w_out,
    const float* __restrict__ b_out, const float* __restrict__ U_re,
    const float* __restrict__ U_im, const float* __restrict__ w_cls,
    const float* __restrict__ b_cls, const float* __restrict__ zc,
    float* __restrict__ out) {
  __shared__ float4 s_img4[196];  // 784 floats, 16B aligned
  __shared__ float s_red1[8];
  __shared__ float s_red2[4];
  const float* s_img = (const float*)s_img4;

  const int b = blockIdx.x;
  const int tid = (int)threadIdx.x;
  const int lane = tid & 31;
  const int wid = tid >> 5;
  const float4* img4 = (const float4*)(x + (size_t)b * 784);

  // ---- Phase 1: stage image (float4) + local sums ----
  float lsum = 0.f, lsq = 0.f;
  for (int i = tid; i < 196; i += 128) {
    float4 v = img4[i];
    s_img4[i] = v;
    lsum += (v.x + v.y) + (v.z + v.w);
    lsq = fmaf(v.x, v.x, fmaf(v.y, v.y, fmaf(v.z, v.z, fmaf(v.w, v.w, lsq))));
  }
#pragma unroll
  for (int m = 16; m; m >>= 1) {
    lsum += __shfl_xor(lsum, m, 32);
    lsq += __shfl_xor(lsq, m, 32);
  }
  if (lane == 0) {
    s_red1[wid] = lsum;
    s_red1[4 + wid] = lsq;
  }
  __syncthreads();  // s_img ready for all; s_red1 ready for thread 0

  float cls_term = 0.f;  // live only in thread 0
  if (tid == 0) {
    float sum = s_red1[0] + s_red1[1] + s_red1[2] + s_red1[3];
    float sq = s_red1[4] + s_red1[5] + s_red1[6] + s_red1[7];
    float mean = sum * (1.f / 784.f);
    float var = (sq - sum * mean) * (1.f / 783.f);  // ddof=1
    float sd = sqrtf(fmaxf(var, 0.f));
    float h0 = mean, h1 = sd;
    float t0 = tanhf(fmaf(h0, w_in[0], fmaf(h1, w_in[1], b_in[0])));
    float t1 = tanhf(fmaf(h0, w_in[2], fmaf(h1, w_in[3], b_in[1])));
    h0 = fmaf(t0, scale_in[0], shift_in[0]);
    h1 = fmaf(t1, scale_in[1], shift_in[1]);
#pragma unroll
    for (int i = 0; i < 2; ++i) {  // L = 2 clipped layers
      const float* W = Wc + 4 * i;
      t0 = tanhf(fmaf(h0, W[0], fmaf(h1, W[1], bc[2 * i + 0])));
      t1 = tanhf(fmaf(h0, W[2], fmaf(h1, W[3], bc[2 * i + 1])));
      h0 = fmaf(t0, scalec[2 * i + 0], shiftc[2 * i + 0]);
      h1 = fmaf(t1, scalec[2 * i + 1], shiftc[2 * i + 1]);
    }
    float cls_out = fmaf(h0, w_out[0], fmaf(h1, w_out[1], b_out[0]));
    cls_term = w_cls[0] * cls_out;
  }

  // ---- Preload U^T into WMMA B-layout registers (once per wave) ----
  // B[k][n] = U[n][k]; B layout: VGPR v, lane l -> K = 4c + v + 2*(l/16),
  // N = l%16 (mirrors 16x4 f32 A layout).
  const int half = lane >> 4;
  const int n = lane & 15;
  v2f Bre[4], Bim[4];
#pragma unroll
  for (int c = 0; c < 4; ++c) {
    int k0 = 4 * c + 2 * half;
    Bre[c][0] = U_re[n * 16 + k0];
    Bre[c][1] = U_re[n * 16 + k0 + 1];
    Bim[c][0] = U_im[n * 16 + k0];
    Bim[c][1] = U_im[n * 16 + k0 + 1];
  }

  // ---- Phase 2: patch tiles.  patch m = lane%16, basis k = q0q1q2q3 with
  // chunk c = q0q1, half = q2, vgpr v = q3.  psi = a0[q0]a1[q1]a2[q2]a3[q3].
  // Tiles: wave w -> {w, w+4, w+8}; wave 3 additionally tile 12 (wave 0
  // carries the MLP, so give the extra tile to wave 3).
  float acc = 0.f;
  const int ntiles = (wid == 3) ? 4 : 3;
  for (int j = 0; j < ntiles; ++j) {  // uniform per wave -> EXEC all 1s
    const int t = (j < 3) ? (wid + 4 * j) : 12;
    const int pbase = t * 16;
    int p = pbase + n;
    int pe = p < 196 ? p : 195;  // clamp (zc rows >=196 are zero)
    int pr = pe / 14, pc = pe - pr * 14;
    const float* pp = &s_img[(2 * pr) * 28 + 2 * pc];
    float x0 = pp[0], x1 = pp[1], x2 = pp[28], x3 = pp[29];
    float c0 = __cosf(0.5f * x0), s0 = __sinf(0.5f * x0);
    float c1 = __cosf(0.5f * x1), s1 = __sinf(0.5f * x1);
    float c2 = __cosf(0.5f * x2), s2 = __sinf(0.5f * x2);
    float c3 = __cosf(0.5f * x3), s3 = __sinf(0.5f * x3);
    float f2 = half ? s2 : c2;
    float h0f = f2 * c3, h1f = f2 * s3;
    float g[4] = {c0 * c1, c0 * s1, s0 * c1, s0 * s1};

    // unconditional, hoistable zc loads (rows pbase+8*half .. +7)
    const float* zrow = zc + (pbase + 8 * half) * 16 + n;
    float zv[8];
#pragma unroll
    for (int r = 0; r < 8; ++r) zv[r] = zrow[r * 16];

    v8f re = {};
    v8f im = {};
#pragma unroll
    for (int c = 0; c < 4; ++c) {
      v2f A;
      A[0] = g[c] * h0f;  // K = 4c + 2*half + 0  (q3=0 -> cos)
      A[1] = g[c] * h1f;  // K = 4c + 2*half + 1  (q3=1 -> sin)
      re = __builtin_amdgcn_wmma_f32_16x16x4_f32(false, A, false, Bre[c],
                                                 (short)0, re, false, false);
      im = __builtin_amdgcn_wmma_f32_16x16x4_f32(false, A, false, Bim[c],
                                                 (short)0, im, false, false);
    }
    // C/D layout: VGPR r, lane -> patch M = pbase + r + 8*half, basis N = n
#pragma unroll
    for (int r = 0; r < 8; ++r) {
      acc = fmaf(fmaf(re[r], re[r], im[r] * im[r]), zv[r], acc);
    }
  }

  // ---- Phase 3: reduce + finalize ----
#pragma unroll
  for (int m = 16; m; m >>= 1) acc += __shfl_xor(acc, m, 32);
  if (lane == 0) s_red2[wid] = acc;
  __syncthreads();
  if (tid == 0) {
    float q = s_red2[0] + s_red2[1] + s_red2[2] + s_red2[3];
    float logit = b_cls[0] + cls_term + q;
    out[b] = 1.f / (1.f + __expf(-logit));
  }
}

// ---------------------------------------------------------------------------
extern "C" void kernel_launch(void* const* d_in, const int* in_sizes, int n_in,
                              void* d_out, int out_size, void* d_ws,
                              size_t ws_size, hipStream_t stream) {
  const float* x = (const float*)d_in[0];
  const float* w_in = (const float*)d_in[1];
  const float* b_in = (const float*)d_in[2];
  const float* scale_in = (const float*)d_in[3];
  const float* shift_in = (const float*)d_in[4];
  const float* Wc = (const float*)d_in[5];
  const float* bc = (const float*)d_in[6];
  const float* scalec = (const float*)d_in[7];
  const float* shiftc = (const float*)d_in[8];
  const float* w_out = (const float*)d_in[9];
  const float* b_out = (const float*)d_in[10];
  const float* U_re = (const float*)d_in[11];
  const float* U_im = (const float*)d_in[12];
  const float* w_cls = (const float*)d_in[13];
  const float* b_cls = (const float*)d_in[14];
  float* zc = (float*)d_ws;  // ZC_ROWS*16 floats = 13312 B
  float* out = (float*)d_out;

  const int B = in_sizes[0] / 784;

  zc_setup_kernel<<<1, 256, 0, stream>>>(w_cls, zc);
  fraud_fused_kernel<<<B, 128, 0, stream>>>(
      x, w_in, b_in, scale_in, shift_in, Wc, bc, scalec, shiftc, w_out, b_out,
      U_re, U_im, w_cls, b_cls, zc, out);
}